// LearnableAlphaEmbeddingSelector_78202764525973
// MI455X (gfx1250) — compile-verified
//
#include <hip/hip_runtime.h>
#include <math.h>

typedef __attribute__((ext_vector_type(2))) float v2f;
typedef __attribute__((ext_vector_type(4))) float v4f;
typedef __attribute__((ext_vector_type(8))) float v8f;

#define B_SZ   4
#define N_SZ   512
#define D_SZ   768
#define V_SZ   50257
#define K_SEL  64
#define VTILES ((V_SZ + 15) / 16)      // 3142
#define LDS_STRIDE 772                 // 768 + 4 pad -> conflict-free B reads

__device__ __forceinline__ float wave_sum(float v) {
#pragma unroll
  for (int off = 16; off > 0; off >>= 1) v += __shfl_xor(v, off, 32);
  return v;
}

__device__ __forceinline__ float sigmoidf(float x) {
  return 1.0f / (1.0f + expf(-x));
}

// ---- K1: per-key inverse L2 norm -------------------------------------------
__global__ void key_norm_kernel(const float* __restrict__ key,
                                float* __restrict__ invk) {
  int row  = blockIdx.x * 8 + (threadIdx.x >> 5);
  int lane = threadIdx.x & 31;
  if (row >= V_SZ) return;
  const float* p = key + (size_t)row * D_SZ;
  float ss = 0.f;
  for (int k = lane; k < D_SZ; k += 32) { float x = p[k]; ss += x * x; }
  ss = wave_sum(ss);
  if (lane == 0) invk[row] = 1.0f / fmaxf(sqrtf(ss), 1e-12f);
}

// ---- K2: normalized query rows ---------------------------------------------
__global__ void query_norm_kernel(const float* __restrict__ q,
                                  float* __restrict__ qn) {
  int row  = blockIdx.x * 8 + (threadIdx.x >> 5);
  int lane = threadIdx.x & 31;
  if (row >= B_SZ * N_SZ) return;
  const float* p = q + (size_t)row * D_SZ;
  float ss = 0.f;
  for (int k = lane; k < D_SZ; k += 32) { float x = p[k]; ss += x * x; }
  ss = wave_sum(ss);
  float inv = 1.0f / fmaxf(sqrtf(ss), 1e-12f);
  float* o = qn + (size_t)row * D_SZ;
  for (int k = lane; k < D_SZ; k += 32) o[k] = p[k] * inv;
}

// ---- K3: graph query = mean over N of normalized query ---------------------
__global__ void graph_query_kernel(const float* __restrict__ qn,
                                   float* __restrict__ gq) {
  int id = blockIdx.x * blockDim.x + threadIdx.x;   // B*D = 3072
  if (id >= B_SZ * D_SZ) return;
  int b = id / D_SZ, d = id - b * D_SZ;
  float s = 0.f;
  for (int n = 0; n < N_SZ; ++n)
    s += qn[((size_t)(b * N_SZ + n)) * D_SZ + d];
  gq[id] = s * (1.0f / N_SZ);
}

// ---- K4: WMMA GEMM + N-maxpool + graph sim + fuse --------------------------
__global__ __launch_bounds__(128)
void fused_sim_kernel(const float* __restrict__ key,
                      const float* __restrict__ qn,
                      const float* __restrict__ gq,
                      const float* __restrict__ invk,
                      const float* __restrict__ alpha_raw,
                      float* __restrict__ out_fused,
                      float* __restrict__ scr_fused) {
  __shared__ float Bt[16 * LDS_STRIDE];     // raw key tile, [n][k], padded
  __shared__ float wave_max[4][16];
  __shared__ float gsim[16];

  const int v0   = blockIdx.x * 16;
  const int b    = blockIdx.y;
  const int tid  = threadIdx.x;
  const int lane = tid & 31;
  const int w    = tid >> 5;

  // cooperative float4 stage of 16 key rows into LDS (zero-pad past V)
  for (int f = tid; f < 16 * (D_SZ / 4); f += 128) {
    int n  = f / (D_SZ / 4);
    int k4 = (f - n * (D_SZ / 4)) * 4;
    int vr = v0 + n;
    v4f val = {};
    if (vr < V_SZ) val = *(const v4f*)(key + (size_t)vr * D_SZ + k4);
    *(v4f*)(&Bt[n * LDS_STRIDE + k4]) = val;
  }
  __syncthreads();

  // ISA 7.12.2 32-bit operand layout: lane l -> row/col (l&15), K pair (l>>4)*2
  const int m  = lane & 15;
  const int bk = (lane >> 4) << 1;
  float runmax = -INFINITY;

  for (int chunk = w; chunk < N_SZ / 16; chunk += 4) {
    const float* arow = qn + ((size_t)(b * N_SZ + chunk * 16 + m)) * D_SZ + bk;
    const float* brow = &Bt[m * LDS_STRIDE + bk];
    v8f acc = {};
#pragma unroll 4
    for (int k0 = 0; k0 < D_SZ; k0 += 4) {
      v2f a  = *(const v2f*)(arow + k0);
      v2f bb = *(const v2f*)(brow + k0);
      acc = __builtin_amdgcn_wmma_f32_16x16x4_f32(
          false, a, false, bb, (short)0, acc, false, false);
    }
    // column max of the 16x16 tile: 8 regs per lane + lane<->lane^16 combine
    float mx = acc[0];
#pragma unroll
    for (int i = 1; i < 8; ++i) mx = fmaxf(mx, acc[i]);
    mx = fmaxf(mx, __shfl_xor(mx, 16, 32));
    runmax = fmaxf(runmax, mx);
  }
  if (lane < 16) wave_max[w][lane] = runmax;
  __syncthreads();

  // graph similarity from the staged raw tile
  if (tid < 16) {
    const float* g  = gq + (size_t)b * D_SZ;
    const float* kr = &Bt[tid * LDS_STRIDE];
    float s = 0.f;
    for (int k = 0; k < D_SZ; ++k) s = fmaf(g[k], kr[k], s);
    gsim[tid] = s;
  }
  __syncthreads();

  if (tid < 16) {
    int v = v0 + tid;
    if (v < V_SZ) {
      float nm = fmaxf(fmaxf(wave_max[0][tid], wave_max[1][tid]),
                       fmaxf(wave_max[2][tid], wave_max[3][tid]));
      float inv   = invk[v];                     // normalize AFTER max (inv>0)
      float alpha = sigmoidf(alpha_raw[0]);
      float fused = alpha * (nm * inv) + (1.0f - alpha) * (gsim[tid] * inv);
      out_fused[(size_t)b * V_SZ + v] = fused;
      scr_fused[(size_t)b * V_SZ + v] = fused;
    }
  }
}

// ---- K5: iterative top-64 per batch (tie -> lower index, like lax.top_k) ---
__global__ void topk_kernel(float* __restrict__ scr,
                            int* __restrict__ idx_i,
                            float* __restrict__ out_idx) {
  __shared__ float sval[256];
  __shared__ int   sidx[256];
  const int b   = blockIdx.x;
  const int tid = threadIdx.x;
  float* f = scr + (size_t)b * V_SZ;
  for (int r = 0; r < K_SEL; ++r) {
    float best = -INFINITY; int bi = V_SZ;
    for (int v = tid; v < V_SZ; v += 256) {
      float x = f[v];
      if (x > best || (x == best && v < bi)) { best = x; bi = v; }
    }
    sval[tid] = best; sidx[tid] = bi;
    __syncthreads();
    for (int s = 128; s > 0; s >>= 1) {
      if (tid < s) {
        float xv = sval[tid + s]; int xi = sidx[tid + s];
        if (xv > sval[tid] || (xv == sval[tid] && xi < sidx[tid])) {
          sval[tid] = xv; sidx[tid] = xi;
        }
      }
      __syncthreads();
    }
    if (tid == 0) {
      int sel = sidx[0];
      idx_i[b * K_SEL + r]   = sel;
      out_idx[b * K_SEL + r] = (float)sel;
      f[sel] = -INFINITY;
    }
    __syncthreads();
  }
}

// ---- K6: gather selected key rows + alpha ----------------------------------
__global__ void gather_kernel(const float* __restrict__ key,
                              const int* __restrict__ idx_i,
                              const float* __restrict__ alpha_raw,
                              float* __restrict__ out_sel,
                              float* __restrict__ out_alpha) {
  int i = blockIdx.x * blockDim.x + threadIdx.x;
  if (i == 0) out_alpha[0] = sigmoidf(alpha_raw[0]);
  if (i < B_SZ * K_SEL * D_SZ) {
    int d   = i % D_SZ;
    int t   = i / D_SZ;             // b*K_SEL + r
    int row = idx_i[t];
    out_sel[i] = key[(size_t)row * D_SZ + d];
  }
}

extern "C" void kernel_launch(void* const* d_in, const int* in_sizes, int n_in,
                              void* d_out, int out_size, void* d_ws, size_t ws_size,
                              hipStream_t stream) {
  (void)in_sizes; (void)n_in; (void)out_size; (void)ws_size;
  const float* q    = (const float*)d_in[0];
  const float* key  = (const float*)d_in[1];
  const float* araw = (const float*)d_in[2];

  // workspace layout (floats): ~7.3 MB total
  float* ws   = (float*)d_ws;
  float* invk = ws;                        // 50257  (pad to 50304)
  float* qn   = ws + 50304;                // 1572864
  float* gq   = qn + 1572864;              // 3072
  float* scr  = gq + 3072;                 // 201028 (pad to 201088)
  int*   idxi = (int*)(scr + 201088);      // 256 ints

  // output layout (floats): sel | idx | fused | alpha
  float* out_sel   = (float*)d_out;                 // 4*64*768 = 196608
  float* out_idx   = out_sel + B_SZ * K_SEL * D_SZ; // 256
  float* out_fused = out_idx + B_SZ * K_SEL;        // 4*50257 = 201028
  float* out_alpha = out_fused + B_SZ * V_SZ;       // 1

  key_norm_kernel<<<(V_SZ + 7) / 8, 256, 0, stream>>>(key, invk);
  query_norm_kernel<<<(B_SZ * N_SZ) / 8, 256, 0, stream>>>(q, qn);
  graph_query_kernel<<<(B_SZ * D_SZ + 255) / 256, 256, 0, stream>>>(qn, gq);

  dim3 g4(VTILES, B_SZ);
  fused_sim_kernel<<<g4, 128, 0, stream>>>(key, qn, gq, invk, araw,
                                           out_fused, scr);

  topk_kernel<<<B_SZ, 256, 0, stream>>>(scr, idxi, out_idx);
  gather_kernel<<<(B_SZ * K_SEL * D_SZ + 255) / 256, 256, 0, stream>>>(
      key, idxi, araw, out_sel, out_alpha);
}